// DataAugmentation_46823733461007
// MI455X (gfx1250) — compile-verified
//
#include <hip/hip_runtime.h>
#include <hip/hip_bf16.h>

// ---------------------------------------------------------------------------
// Data augmentation pipeline for MI455X (gfx1250).
//   x:(256,3,224,224) f32  -> flip -> +0.625*noise -> *(0.85+0.3*bright)
//   -> perspective warp (bilinear, zero pad)
//   -> rotate (bilinear, zero pad)
//   -> crop 190x190 at crop_ij -> bicubic resize to 224x224
// Bicubic resize is separable => two f32 GEMMs per (b,c) plane, executed with
// V_WMMA_F32_16X16X4_F32. B operand K-slices are streamed to LDS with
// GLOBAL_LOAD_ASYNC_TO_LDS_B128 (ASYNCcnt) and consumed via ds loads.
// ---------------------------------------------------------------------------

typedef __attribute__((ext_vector_type(2))) float v2f;
typedef __attribute__((ext_vector_type(8))) float v8f;

#define BN 256
#define CN 3
#define HN 224
#define WN 224
#define CROPN 190
#define KPAD 192            // K dim padded to multiple of 16
#define PLANES (BN * CN)    // 768
#define PLANE_ELEMS (HN * WN)

// ---------------------------------------------------------------------------
// Kernel 0: per-batch perspective coefficients (8x8 Gauss-Jordan solve).
// ---------------------------------------------------------------------------
__global__ void __launch_bounds__(256) persp_coeffs_kernel(
    const int* __restrict__ ep_raw, float* __restrict__ coeffs) {
  int b = blockIdx.x * blockDim.x + threadIdx.x;
  if (b >= BN) return;

  const float sxv[4] = {0.f, 223.f, 223.f, 0.f};
  const float syv[4] = {0.f, 0.f, 223.f, 223.f};
  const float ofx[4] = {0.f, 195.f, 195.f, 0.f};
  const float ofy[4] = {0.f, 0.f, 195.f, 195.f};

  float M[8][9];
  for (int i = 0; i < 4; ++i) {
    float ex = (float)ep_raw[b * 8 + i * 2 + 0] + ofx[i];
    float ey = (float)ep_raw[b * 8 + i * 2 + 1] + ofy[i];
    float sx = sxv[i], sy = syv[i];
    M[i][0] = ex;  M[i][1] = ey;  M[i][2] = 1.f;
    M[i][3] = 0.f; M[i][4] = 0.f; M[i][5] = 0.f;
    M[i][6] = -sx * ex; M[i][7] = -sx * ey; M[i][8] = sx;
    M[i + 4][0] = 0.f; M[i + 4][1] = 0.f; M[i + 4][2] = 0.f;
    M[i + 4][3] = ex;  M[i + 4][4] = ey;  M[i + 4][5] = 1.f;
    M[i + 4][6] = -sy * ex; M[i + 4][7] = -sy * ey; M[i + 4][8] = sy;
  }

  for (int col = 0; col < 8; ++col) {
    int piv = col;
    float best = fabsf(M[col][col]);
    for (int r = col + 1; r < 8; ++r) {
      float v = fabsf(M[r][col]);
      if (v > best) { best = v; piv = r; }
    }
    if (piv != col) {
      for (int j = 0; j < 9; ++j) {
        float t = M[col][j]; M[col][j] = M[piv][j]; M[piv][j] = t;
      }
    }
    float inv = 1.f / M[col][col];
    for (int j = col; j < 9; ++j) M[col][j] *= inv;
    for (int r = 0; r < 8; ++r) {
      if (r == col) continue;
      float f = M[r][col];
      if (f != 0.f)
        for (int j = col; j < 9; ++j) M[r][j] -= f * M[col][j];
    }
  }
  for (int i = 0; i < 8; ++i) coeffs[b * 8 + i] = M[i][8];
}

// ---------------------------------------------------------------------------
// Kernel 1: bicubic (Keys a=-0.5, jax renormalized) weight matrices.
//   Wht: (KPAD x 224)  Wht[p][u]  (transposed horizontal weights)
//   Wv : (224 x KPAD)  Wv[v][r]
// Pad rows/cols (190,191) are zero so the K=192 padded GEMM is exact.
// ---------------------------------------------------------------------------
__device__ __forceinline__ float cubw(float t) {
  t = fabsf(t);
  float t2 = t * t;
  if (t <= 1.f) return (1.5f * t - 2.5f) * t2 + 1.f;
  if (t < 2.f)  return ((-0.5f * t + 2.5f) * t - 4.f) * t + 2.f;
  return 0.f;
}

__device__ __forceinline__ float resize_weight(int o, int i) {
  const float ratio = (float)CROPN / (float)HN;   // 190/224 (upsample)
  float s = ((float)o + 0.5f) * ratio - 0.5f;
  int k0 = (int)floorf(s) - 1;
  float sum = 0.f, w = 0.f;
  for (int t = 0; t < 4; ++t) {
    int q = k0 + t;
    if (q >= 0 && q < CROPN) {
      float wv = cubw(s - (float)q);
      sum += wv;
      if (q == i) w = wv;
    }
  }
  return (sum != 0.f) ? w / sum : 0.f;
}

__global__ void __launch_bounds__(256) weights_kernel(
    float* __restrict__ Wht, float* __restrict__ Wv) {
  int id = blockIdx.x * blockDim.x + threadIdx.x;
  const int n1 = KPAD * HN;
  if (id < n1) {
    int p = id / HN, u = id % HN;
    Wht[p * HN + u] = (p < CROPN) ? resize_weight(u, p) : 0.f;
  } else if (id < 2 * n1) {
    int id2 = id - n1;
    int v = id2 / KPAD, r = id2 % KPAD;
    Wv[v * KPAD + r] = (r < CROPN) ? resize_weight(v, r) : 0.f;
  }
}

// ---------------------------------------------------------------------------
// Kernel 2: fused flip + noise + brightness + perspective warp.
// ---------------------------------------------------------------------------
__global__ void __launch_bounds__(WN) persp_warp_kernel(
    const float* __restrict__ x, const float* __restrict__ noise,
    const float* __restrict__ brightness, const int* __restrict__ flip_mask,
    const float* __restrict__ coeffs, float* __restrict__ out) {
  const int xo = threadIdx.x;
  const int y  = blockIdx.x;
  const int c  = blockIdx.y;
  const int b  = blockIdx.z;

  const float* cf = coeffs + b * 8;
  const float ca = cf[0], cb = cf[1], cc = cf[2], cd = cf[3];
  const float ce = cf[4], cff = cf[5], cg = cf[6], ch = cf[7];
  const float scale = 0.85f + 0.3f * brightness[b];
  const int flip = flip_mask[b];

  const float Xg = (float)xo + 0.5f;
  const float Yg = (float)y + 0.5f;
  const float den = cg * Xg + ch * Yg + 1.0f;
  const float sx = (ca * Xg + cb * Yg + cc) / den - 0.5f;
  const float sy = (cd * Xg + ce * Yg + cff) / den - 0.5f;

  const size_t base = (size_t)(b * CN + c) * PLANE_ELEMS;
  const float* xp = x + base;
  const float* np = noise + base;

  const float x0 = floorf(sx), y0 = floorf(sy);
  const float wx = sx - x0, wy = sy - y0;
  const int xi = (int)x0, yi = (int)y0;

  float acc = 0.f;
#pragma unroll
  for (int dy = 0; dy < 2; ++dy) {
#pragma unroll
    for (int dx = 0; dx < 2; ++dx) {
      const int xx = xi + dx, yy = yi + dy;
      const bool valid = (xx >= 0) & (xx < WN) & (yy >= 0) & (yy < HN);
      const int xc = min(max(xx, 0), WN - 1);
      const int yc = min(max(yy, 0), HN - 1);
      const int col = flip ? (WN - 1 - xc) : xc;
      const float v = (xp[yc * WN + col] + 0.625f * np[yc * WN + col]) * scale;
      const float w = (dx ? wx : 1.f - wx) * (dy ? wy : 1.f - wy);
      acc += valid ? v * w : 0.f;
    }
  }
  out[base + y * WN + xo] = acc;
}

// ---------------------------------------------------------------------------
// Kernel 3: rotation warp (bilinear, zero padded).
// ---------------------------------------------------------------------------
__global__ void __launch_bounds__(WN) rotate_kernel(
    const float* __restrict__ src, const int* __restrict__ angles,
    float* __restrict__ out) {
  const int xo = threadIdx.x;
  const int y  = blockIdx.x;
  const int c  = blockIdx.y;
  const int b  = blockIdx.z;

  const float th = ((float)angles[b] - 16.f) * (3.14159265358979323846f / 180.f);
  const float cs = cosf(th), sn = sinf(th);
  const float cx = (WN - 1) * 0.5f, cy = (HN - 1) * 0.5f;
  const float dx = (float)xo - cx;
  const float dy = (float)y - cy;
  const float rx = cs * dx + sn * dy + cx;
  const float ry = -sn * dx + cs * dy + cy;

  const size_t base = (size_t)(b * CN + c) * PLANE_ELEMS;
  const float* p = src + base;

  const float x0 = floorf(rx), y0 = floorf(ry);
  const float wx = rx - x0, wy = ry - y0;
  const int xi = (int)x0, yi = (int)y0;

  float acc = 0.f;
#pragma unroll
  for (int dyy = 0; dyy < 2; ++dyy) {
#pragma unroll
    for (int dxx = 0; dxx < 2; ++dxx) {
      const int xx = xi + dxx, yy = yi + dyy;
      const bool valid = (xx >= 0) & (xx < WN) & (yy >= 0) & (yy < HN);
      const int xc = min(max(xx, 0), WN - 1);
      const int yc = min(max(yy, 0), HN - 1);
      const float w = (dxx ? wx : 1.f - wx) * (dyy ? wy : 1.f - wy);
      acc += valid ? p[yc * WN + xc] * w : 0.f;
    }
  }
  out[base + y * WN + xo] = acc;
}

// ---------------------------------------------------------------------------
// Async global->LDS helpers (gfx1250, ASYNCcnt).
//   global_load_async_to_lds_b128: per lane, 16 bytes from global address vaddr to LDS address vdst
// LDS byte address = low 32 bits of the generic pointer (flat LDS aperture
// maps addr[31:0] directly to the LDS offset).
// ---------------------------------------------------------------------------
__device__ __forceinline__ void async_copy_b128(unsigned lds_addr,
                                                const float* gsrc) {
  asm volatile("global_load_async_to_lds_b128 %0, %1, off"
               :: "v"(lds_addr), "v"((unsigned long long)(uintptr_t)gsrc)
               : "memory");
}
__device__ __forceinline__ void wait_async0() {
  asm volatile("s_wait_asynccnt 0x0" ::: "memory");
}

// ---------------------------------------------------------------------------
// WMMA fragment layout (wave32, V_WMMA_F32_16X16X4_F32):
//   A 16x4 f32 : a[j]  = A[lane%16][kbase + 2*(lane/16) + j]
//   B 4x16 f32 : b[j]  = B[kbase + 2*(lane/16) + j][lane%16]
//   C/D 16x16  : c[r]  = C[r + 8*(lane/16)][lane%16]
// Each wave computes a 16x32 strip (two accumulators share one A fragment).
// The block cooperatively stages each 16x224 B K-slice (14 KB) in LDS via
// async b128 copies, then all 8 waves consume it through ds loads.
// ---------------------------------------------------------------------------

#define KSLICE_FLOATS (16 * WN)          // 3584 floats = 14336 bytes
#define KSLICE_VEC4   (KSLICE_FLOATS / 4)  // 896 float4 chunks

// Kernel 4: crop + horizontal bicubic resize as GEMM:
//   tmp[r, u] = sum_p src[i0+r, j0+p] * Wht[p][u]    (per plane)
__global__ void __launch_bounds__(256) resize_h_kernel(
    const float* __restrict__ src, const float* __restrict__ Wht,
    const int* __restrict__ crop_ij, float* __restrict__ tmp) {
  __shared__ __align__(16) float Bs[KSLICE_FLOATS];

  const int plane = blockIdx.x;
  const int b = plane / CN;
  const int i0 = crop_ij[b * 2 + 0];
  const int j0 = crop_ij[b * 2 + 1];

  const int tid   = threadIdx.x;
  const int lane  = tid & 31;
  const int wid   = tid >> 5;
  const int task  = blockIdx.y * 8 + wid;       // (mt, npair): 12*7 = 84 tasks
  const bool valid = task < 12 * 7;
  const int mt = valid ? task / 7 : 0;
  const int np = valid ? task % 7 : 0;

  const int l16   = lane & 15;
  const int khalf = lane >> 4;

  const float* img = src + (size_t)plane * PLANE_ELEMS;
  const int mrow   = mt * 16 + l16;
  const int srcRow = i0 + min(mrow, CROPN - 1);
  const float* arow = img + srcRow * WN + j0;
  const int ncol0 = np * 32 + l16;
  const int ncol1 = ncol0 + 16;

  const unsigned bs_base = (unsigned)(uintptr_t)(void*)Bs;

  v8f acc0 = 0.f, acc1 = 0.f;
  for (int kt = 0; kt < KPAD / 16; ++kt) {
    // Stage Wht rows [kt*16, kt*16+16) x 224 cols into LDS (coalesced b128).
    const float* slice = Wht + kt * KSLICE_FLOATS;
    for (int i = tid; i < KSLICE_VEC4; i += 256)
      async_copy_b128(bs_base + i * 16, slice + i * 4);
    wait_async0();
    __syncthreads();

    if (valid) {
#pragma unroll
      for (int k4 = 0; k4 < 4; ++k4) {
        const int ka = kt * 16 + k4 * 4 + 2 * khalf;   // global k index
        const int lr = k4 * 4 + 2 * khalf;             // row within LDS slice
        const int p0 = min(ka, CROPN - 1);             // pad cols: weight = 0
        const int p1 = min(ka + 1, CROPN - 1);
        v2f a;  a.x = arow[p0];                  a.y = arow[p1];
        v2f b0; b0.x = Bs[lr * WN + ncol0];      b0.y = Bs[(lr + 1) * WN + ncol0];
        v2f b1; b1.x = Bs[lr * WN + ncol1];      b1.y = Bs[(lr + 1) * WN + ncol1];
        acc0 = __builtin_amdgcn_wmma_f32_16x16x4_f32(
            false, a, false, b0, (short)0, acc0, false, false);
        acc1 = __builtin_amdgcn_wmma_f32_16x16x4_f32(
            false, a, false, b1, (short)0, acc1, false, false);
      }
      __builtin_prefetch(arow + min(kt * 16 + 16, CROPN - 1), 0, 0);
    }
    __syncthreads();
  }

  if (valid) {
    float* dst = tmp + (size_t)plane * (KPAD * WN);
    const int hi = khalf * 8;
#pragma unroll
    for (int r = 0; r < 8; ++r) {
      dst[(mt * 16 + r + hi) * WN + np * 32 + l16]      = acc0[r];
      dst[(mt * 16 + r + hi) * WN + np * 32 + 16 + l16] = acc1[r];
    }
  }
}

// Kernel 5: vertical bicubic resize as GEMM:
//   out[v, u] = sum_r Wv[v][r] * tmp[r, u]    (per plane)
__global__ void __launch_bounds__(256) resize_v_kernel(
    const float* __restrict__ tmp, const float* __restrict__ Wv,
    float* __restrict__ out) {
  __shared__ __align__(16) float Bs[KSLICE_FLOATS];

  const int plane = blockIdx.x;
  const int tid   = threadIdx.x;
  const int lane  = tid & 31;
  const int wid   = tid >> 5;
  const int task  = blockIdx.y * 8 + wid;       // (mt, npair): 14*7 = 98 tasks
  const bool valid = task < 14 * 7;
  const int mt = valid ? task / 7 : 0;
  const int np = valid ? task % 7 : 0;

  const int l16   = lane & 15;
  const int khalf = lane >> 4;

  const float* bp  = tmp + (size_t)plane * (KPAD * WN);
  const int vrow   = mt * 16 + l16;             // output row < 224
  const float* wrow = Wv + vrow * KPAD;
  const int ncol0 = np * 32 + l16;
  const int ncol1 = ncol0 + 16;

  const unsigned bs_base = (unsigned)(uintptr_t)(void*)Bs;

  v8f acc0 = 0.f, acc1 = 0.f;
  for (int kt = 0; kt < KPAD / 16; ++kt) {
    // Stage tmp rows [kt*16, kt*16+16) x 224 cols into LDS (coalesced b128).
    const float* slice = bp + kt * KSLICE_FLOATS;
    for (int i = tid; i < KSLICE_VEC4; i += 256)
      async_copy_b128(bs_base + i * 16, slice + i * 4);
    wait_async0();
    __syncthreads();

    if (valid) {
#pragma unroll
      for (int k4 = 0; k4 < 4; ++k4) {
        const int ka = kt * 16 + k4 * 4 + 2 * khalf;
        const int lr = k4 * 4 + 2 * khalf;
        v2f a;  a.x = wrow[ka];                  a.y = wrow[ka + 1];
        v2f b0; b0.x = Bs[lr * WN + ncol0];      b0.y = Bs[(lr + 1) * WN + ncol0];
        v2f b1; b1.x = Bs[lr * WN + ncol1];      b1.y = Bs[(lr + 1) * WN + ncol1];
        acc0 = __builtin_amdgcn_wmma_f32_16x16x4_f32(
            false, a, false, b0, (short)0, acc0, false, false);
        acc1 = __builtin_amdgcn_wmma_f32_16x16x4_f32(
            false, a, false, b1, (short)0, acc1, false, false);
      }
      __builtin_prefetch(wrow + min(kt * 16 + 16, KPAD - 1), 0, 0);
    }
    __syncthreads();
  }

  if (valid) {
    float* dst = out + (size_t)plane * PLANE_ELEMS;
    const int hi = khalf * 8;
#pragma unroll
    for (int r = 0; r < 8; ++r) {
      dst[(mt * 16 + r + hi) * WN + np * 32 + l16]      = acc0[r];
      dst[(mt * 16 + r + hi) * WN + np * 32 + 16 + l16] = acc1[r];
    }
  }
}

// ---------------------------------------------------------------------------
// Workspace layout (bytes):
//   coeffs : 256*8*4                       =      8192   @ 0
//   Wht    : 192*224*4                     =    172032   @ 8192
//   Wv     : 224*192*4                     =    172032   @ 180224
//   warped1: 768*224*224*4                 = 154140672   @ 352256
//   warped2: 768*224*224*4                 = 154140672   @ 154492928
//   tmp    : 768*192*224*4 (aliases warped1, free after rotate)
// Total: ~295 MB.
// ---------------------------------------------------------------------------
extern "C" void kernel_launch(void* const* d_in, const int* in_sizes, int n_in,
                              void* d_out, int out_size, void* d_ws,
                              size_t ws_size, hipStream_t stream) {
  (void)in_sizes; (void)n_in; (void)out_size; (void)ws_size;

  const float* x      = (const float*)d_in[0];
  const float* noise  = (const float*)d_in[1];
  const float* bright = (const float*)d_in[2];
  const int*   flip   = (const int*)d_in[3];
  const int*   ep     = (const int*)d_in[4];
  const int*   ang    = (const int*)d_in[5];
  const int*   crop   = (const int*)d_in[6];
  float*       out    = (float*)d_out;

  char* ws = (char*)d_ws;
  float* coeffs  = (float*)(ws + 0);
  float* Wht     = (float*)(ws + 8192);
  float* Wv      = (float*)(ws + 180224);
  float* warped1 = (float*)(ws + 352256);
  float* warped2 = (float*)(ws + 154492928ULL);
  float* tmp     = warped1;   // reuse after rotate consumes warped1

  persp_coeffs_kernel<<<1, 256, 0, stream>>>(ep, coeffs);
  weights_kernel<<<(2 * KPAD * HN + 255) / 256, 256, 0, stream>>>(Wht, Wv);
  persp_warp_kernel<<<dim3(HN, CN, BN), WN, 0, stream>>>(
      x, noise, bright, flip, coeffs, warped1);
  rotate_kernel<<<dim3(HN, CN, BN), WN, 0, stream>>>(warped1, ang, warped2);
  resize_h_kernel<<<dim3(PLANES, 11), 256, 0, stream>>>(warped2, Wht, crop, tmp);
  resize_v_kernel<<<dim3(PLANES, 13), 256, 0, stream>>>(tmp, Wv, out);
}